// SMPL_37512244364084
// MI455X (gfx1250) — compile-verified
//
#include <hip/hip_runtime.h>
#include <hip/hip_bf16.h>
#include <math.h>

#define BATCH  256
#define NVERT  6890
#define NJOINT 24
#define NBETA  10
#define NPOSE  207
#define NJREG  43
#define COLS   (NVERT * 3)            /* 20670 flattened (V,3) columns */
#define NT_TILES ((COLS + 15) / 16)   /* 1292 N tiles */

typedef __attribute__((ext_vector_type(2))) float v2f;
typedef __attribute__((ext_vector_type(8))) float v8f;

__constant__ int c_parents[NJOINT] = {0,0,0,0,1,2,3,4,5,6,7,8,9,9,9,12,13,14,16,17,18,19,20,21};

// ---------------------------------------------------------------------------
// Kernel A: batch Rodrigues -> Rs (B,24,9) and pose_feature (B,207)
// ---------------------------------------------------------------------------
__global__ void __launch_bounds__(256)
k_rodrigues(const float* __restrict__ theta, float* __restrict__ Rs,
            float* __restrict__ pf) {
  int idx = blockIdx.x * blockDim.x + threadIdx.x;
  if (idx >= BATCH * NJOINT) return;
  int b = idx / NJOINT, j = idx % NJOINT;
  const float* t = theta + (size_t)b * NJOINT * 3 + j * 3;
  float tx = t[0], ty = t[1], tz = t[2];
  float ax = tx + 1e-8f, ay = ty + 1e-8f, az = tz + 1e-8f;
  float angle = sqrtf(ax * ax + ay * ay + az * az);
  float inva = 1.0f / angle;
  float rx = tx * inva, ry = ty * inva, rz = tz * inva;
  float h = 0.5f * angle;
  float qw = cosf(h), s = sinf(h);
  float qx = s * rx, qy = s * ry, qz = s * rz;
  float qn = rsqrtf(qw * qw + qx * qx + qy * qy + qz * qz);
  qw *= qn; qx *= qn; qy *= qn; qz *= qn;
  float R[9];
  R[0] = 1.f - 2.f * (qy * qy + qz * qz);
  R[1] = 2.f * (qx * qy - qw * qz);
  R[2] = 2.f * (qx * qz + qw * qy);
  R[3] = 2.f * (qx * qy + qw * qz);
  R[4] = 1.f - 2.f * (qx * qx + qz * qz);
  R[5] = 2.f * (qy * qz - qw * qx);
  R[6] = 2.f * (qx * qz - qw * qy);
  R[7] = 2.f * (qy * qz + qw * qx);
  R[8] = 1.f - 2.f * (qx * qx + qy * qy);
  float* rdst = Rs + ((size_t)b * NJOINT + j) * 9;
#pragma unroll
  for (int i = 0; i < 9; i++) rdst[i] = R[i];
  if (j >= 1) {
    float* p = pf + (size_t)b * NPOSE + (j - 1) * 9;
#pragma unroll
    for (int i = 0; i < 9; i++)
      p[i] = R[i] - ((i == 0 || i == 4 || i == 8) ? 1.f : 0.f);
  }
}

// ---------------------------------------------------------------------------
// Kernel B: fused blend-shape GEMMs via V_WMMA_F32_16X16X4_F32.
//   v_shaped = beta @ shapedirs + v_template       (K=10)
//   v_posed  = v_shaped + pose_feature @ posedirs  (K=207)
// Block = 512 threads = 16 waves; wave w owns M-tile w (batch rows w*16..),
// all 16 waves share the same 16 N columns -> posedirs read ~once from HBM.
// Hot K-loops are branch-free; only the final k-step of each phase uses a
// clamped-address load scaled by a {0,1} mask (no exec-mask branches, EXEC
// stays all-1s as WMMA requires).
// Layouts per CDNA5 ISA 7.12.2 (f32 16x16x4):
//   A: lanes 0-15 hold M=lane K={0,1}; lanes 16-31 hold M=lane-16 K={2,3}
//   B: lanes 0-15 hold N=lane K-rows {0,1}; lanes 16-31 N=lane-16 rows {2,3}
//   C/D: VGPR r -> row r (lanes 0-15) / row r+8 (lanes 16-31), N = lane%16
// ---------------------------------------------------------------------------
__global__ void __launch_bounds__(512)
k_blend(const float* __restrict__ beta, const float* __restrict__ pf,
        const float* __restrict__ shapedirs, const float* __restrict__ posedirs,
        const float* __restrict__ v_template,
        float* __restrict__ vs, float* __restrict__ vp) {
  const int lane = threadIdx.x & 31;
  const int wave = threadIdx.x >> 5;   // 0..15 -> M tile
  const int lrow = lane & 15;
  const int half = lane >> 4;
  const int m0 = wave * 16;
  const int n  = blockIdx.x * 16 + lrow;
  const int nc = n < COLS ? n : COLS - 1;   // clamp loads on boundary tile

  v8f c = {0.f, 0.f, 0.f, 0.f, 0.f, 0.f, 0.f, 0.f};

  // ---------------- Phase 1: beta (256x10) @ shapedirs (10x20670) ----------
  {
    const float* arow = beta + (size_t)(m0 + lrow) * NBETA;
    const float* bcol = shapedirs + nc;
    // k0 = 0, 4 : all kk in range (max 7 < 10), branch-free
#pragma unroll
    for (int k0 = 0; k0 < 8; k0 += 4) {
      int kk = k0 + half * 2;
      v2f a, bb;
      a.x  = arow[kk];
      a.y  = arow[kk + 1];
      bb.x = bcol[(size_t)kk * COLS];
      bb.y = bcol[(size_t)(kk + 1) * COLS];
      c = __builtin_amdgcn_wmma_f32_16x16x4_f32(false, a, false, bb,
                                                (short)0, c, false, false);
    }
    // tail k0 = 8 : kk = 8,9 valid; 10,11 masked to zero (clamped load * 0)
    {
      int kk = 8 + half * 2;
      int k0c = kk     < NBETA ? kk     : NBETA - 1;
      int k1c = kk + 1 < NBETA ? kk + 1 : NBETA - 1;
      float msk0 = kk     < NBETA ? 1.f : 0.f;
      float msk1 = kk + 1 < NBETA ? 1.f : 0.f;
      v2f a, bb;
      a.x  = arow[k0c] * msk0;
      a.y  = arow[k1c] * msk1;
      bb.x = bcol[(size_t)k0c * COLS];
      bb.y = bcol[(size_t)k1c * COLS];
      c = __builtin_amdgcn_wmma_f32_16x16x4_f32(false, a, false, bb,
                                                (short)0, c, false, false);
    }
  }
  float vt = v_template[nc];
#pragma unroll
  for (int r = 0; r < 8; r++) {
    int m = m0 + r + half * 8;
    if (n < COLS) vs[(size_t)m * COLS + n] = c[r] + vt;
  }

  // -------- Phase 2: pose_feature (256x207) @ posedirs (207x20670) ---------
  {
    const float* arow = pf + (size_t)(m0 + lrow) * NPOSE;
    const float* bcol = posedirs + nc;
    // k0 = 0..200 : kk max = 203 < 207, fully branch-free (51 iterations)
#pragma unroll 3
    for (int k0 = 0; k0 < 204; k0 += 4) {
      int kk = k0 + half * 2;
      v2f a, bb;
      a.x  = arow[kk];
      a.y  = arow[kk + 1];
      bb.x = bcol[(size_t)kk * COLS];
      bb.y = bcol[(size_t)(kk + 1) * COLS];
      c = __builtin_amdgcn_wmma_f32_16x16x4_f32(false, a, false, bb,
                                                (short)0, c, false, false);
    }
    // tail k0 = 204 : kk = 204..206 valid; 207 masked to zero
    {
      int kk = 204 + half * 2;
      int k0c = kk     < NPOSE ? kk     : NPOSE - 1;
      int k1c = kk + 1 < NPOSE ? kk + 1 : NPOSE - 1;
      float msk0 = kk     < NPOSE ? 1.f : 0.f;
      float msk1 = kk + 1 < NPOSE ? 1.f : 0.f;
      v2f a, bb;
      a.x  = arow[k0c] * msk0;
      a.y  = arow[k1c] * msk1;
      bb.x = bcol[(size_t)k0c * COLS];
      bb.y = bcol[(size_t)k1c * COLS];
      c = __builtin_amdgcn_wmma_f32_16x16x4_f32(false, a, false, bb,
                                                (short)0, c, false, false);
    }
  }
#pragma unroll
  for (int r = 0; r < 8; r++) {
    int m = m0 + r + half * 8;
    if (n < COLS) vp[(size_t)m * COLS + n] = c[r] + vt;
  }
}

// ---------------------------------------------------------------------------
// Kernel C/F: regressor @ vertices per batch.  blockIdx.x = b*nrows + j.
// Sources stay L2-resident (v_shaped / verts each ~21 MB << 192 MB L2).
// ---------------------------------------------------------------------------
__global__ void __launch_bounds__(256)
k_jreg(const float* __restrict__ reg, const float* __restrict__ verts,
       float* __restrict__ Jout, int nrows) {
  int b = blockIdx.x / nrows;
  int j = blockIdx.x % nrows;
  const float* r = reg + (size_t)j * NVERT;
  const float* v = verts + (size_t)b * COLS;
  float ax = 0.f, ay = 0.f, az = 0.f;
  for (int i = threadIdx.x; i < NVERT; i += 256) {
    float w = r[i];
    ax += w * v[3 * i + 0];
    ay += w * v[3 * i + 1];
    az += w * v[3 * i + 2];
  }
  __shared__ float sx[256], sy[256], sz[256];
  sx[threadIdx.x] = ax; sy[threadIdx.x] = ay; sz[threadIdx.x] = az;
  __syncthreads();
  for (int s = 128; s > 0; s >>= 1) {
    if (threadIdx.x < s) {
      sx[threadIdx.x] += sx[threadIdx.x + s];
      sy[threadIdx.x] += sy[threadIdx.x + s];
      sz[threadIdx.x] += sz[threadIdx.x + s];
    }
    __syncthreads();
  }
  if (threadIdx.x == 0) {
    float* o = Jout + ((size_t)b * nrows + j) * 3;
    o[0] = sx[0]; o[1] = sy[0]; o[2] = sz[0];
  }
}

// ---------------------------------------------------------------------------
// Kernel D: serial kinematic chain, one thread per batch, chain state in LDS.
// Stores A as (B,24,12) = rows 0..2 of the 4x4 (row 3 is [0,0,0,1] and the
// weights sum to 1, so row 3 of T is always [0,0,0,1]).
// ---------------------------------------------------------------------------
__global__ void __launch_bounds__(32)
k_kinematics(const float* __restrict__ Rs, const float* __restrict__ J,
             float* __restrict__ Amat) {
  __shared__ float res[32][NJOINT * 12];
  int b = blockIdx.x * 32 + threadIdx.x;
  if (b >= BATCH) return;
  float* W = res[threadIdx.x];
  const float* Rb = Rs + (size_t)b * NJOINT * 9;
  const float* Jb = J + (size_t)b * NJOINT * 3;
  // root: root_R = Rs[0] @ diag(1,-1,-1)  (negate cols 1 and 2)
  {
    const float* R = Rb;
    W[0] = R[0]; W[1] = -R[1]; W[2]  = -R[2]; W[3]  = Jb[0];
    W[4] = R[3]; W[5] = -R[4]; W[6]  = -R[5]; W[7]  = Jb[1];
    W[8] = R[6]; W[9] = -R[7]; W[10] = -R[8]; W[11] = Jb[2];
  }
  for (int i = 1; i < NJOINT; i++) {
    int p = c_parents[i];
    const float* R = Rb + i * 9;
    float tx = Jb[i * 3 + 0] - Jb[p * 3 + 0];
    float ty = Jb[i * 3 + 1] - Jb[p * 3 + 1];
    float tz = Jb[i * 3 + 2] - Jb[p * 3 + 2];
    float L[12] = {R[0], R[1], R[2], tx,
                   R[3], R[4], R[5], ty,
                   R[6], R[7], R[8], tz};
    const float* P = W + p * 12;
    float* O = W + i * 12;
#pragma unroll
    for (int r = 0; r < 3; r++) {
#pragma unroll
      for (int cI = 0; cI < 4; cI++) {
        float acc = P[r * 4 + 0] * L[0 * 4 + cI] +
                    P[r * 4 + 1] * L[1 * 4 + cI] +
                    P[r * 4 + 2] * L[2 * 4 + cI];
        if (cI == 3) acc += P[r * 4 + 3];
        O[r * 4 + cI] = acc;
      }
    }
  }
  // A = results, with column 3 minus results[:, :3] @ J[i]
  float* Aout = Amat + (size_t)b * NJOINT * 12;
  for (int i = 0; i < NJOINT; i++) {
    const float* P = W + i * 12;
    float jx = Jb[i * 3], jy = Jb[i * 3 + 1], jz = Jb[i * 3 + 2];
#pragma unroll
    for (int r = 0; r < 3; r++) {
      Aout[i * 12 + r * 4 + 0] = P[r * 4 + 0];
      Aout[i * 12 + r * 4 + 1] = P[r * 4 + 1];
      Aout[i * 12 + r * 4 + 2] = P[r * 4 + 2];
      Aout[i * 12 + r * 4 + 3] = P[r * 4 + 3] -
          (P[r * 4 + 0] * jx + P[r * 4 + 1] * jy + P[r * 4 + 2] * jz);
    }
  }
}

// ---------------------------------------------------------------------------
// Kernel E: LBS skinning — streams the 169 MB weights tensor exactly once.
// verts[b,v] = sum_j w[b,v,j] * (A[b,j] @ [v_posed,1])
// ---------------------------------------------------------------------------
__global__ void __launch_bounds__(256)
k_skin(const float* __restrict__ weights, const float* __restrict__ Amat,
       const float* __restrict__ vp, float* __restrict__ verts) {
  int b = blockIdx.y;
  __shared__ float sA[NJOINT * 12];
  for (int i = threadIdx.x; i < NJOINT * 12; i += 256)
    sA[i] = Amat[(size_t)b * NJOINT * 12 + i];
  __syncthreads();
  int v = blockIdx.x * 256 + threadIdx.x;
  if (v >= NVERT) return;
  const float* pv = vp + (size_t)b * COLS + 3 * v;
  float x = pv[0], y = pv[1], z = pv[2];
  const float* wv = weights + ((size_t)b * NVERT + v) * NJOINT;
  float a0 = 0.f, a1 = 0.f, a2 = 0.f;
#pragma unroll 4
  for (int j = 0; j < NJOINT; j++) {
    float w = wv[j];
    const float* A = sA + j * 12;
    a0 += w * (A[0] * x + A[1] * y + A[2]  * z + A[3]);
    a1 += w * (A[4] * x + A[5] * y + A[6]  * z + A[7]);
    a2 += w * (A[8] * x + A[9] * y + A[10] * z + A[11]);
  }
  float* o = verts + (size_t)b * COLS + 3 * v;
  o[0] = a0; o[1] = a1; o[2] = a2;
}

// ---------------------------------------------------------------------------
extern "C" void kernel_launch(void* const* d_in, const int* in_sizes, int n_in,
                              void* d_out, int out_size, void* d_ws,
                              size_t ws_size, hipStream_t stream) {
  const float* beta            = (const float*)d_in[0];
  const float* theta           = (const float*)d_in[1];
  const float* v_template      = (const float*)d_in[2];
  const float* shapedirs       = (const float*)d_in[3];
  const float* posedirs        = (const float*)d_in[4];
  const float* J_regressor     = (const float*)d_in[5];
  const float* joint_regressor = (const float*)d_in[6];
  const float* weights         = (const float*)d_in[7];
  float* out = (float*)d_out;
  float* ws  = (float*)d_ws;

  // Workspace layout (floats): total ~10.78M floats (~43.1 MB)
  float* Rs   = ws;                                   // 256*24*9
  float* pf   = Rs + (size_t)BATCH * NJOINT * 9;      // 256*207
  float* vs   = pf + (size_t)BATCH * NPOSE;           // 256*20670
  float* vp   = vs + (size_t)BATCH * COLS;            // 256*20670
  float* Jb   = vp + (size_t)BATCH * COLS;            // 256*24*3
  float* Amat = Jb + (size_t)BATCH * NJOINT * 3;      // 256*24*12

  k_rodrigues<<<(BATCH * NJOINT + 255) / 256, 256, 0, stream>>>(theta, Rs, pf);
  k_blend<<<NT_TILES, 512, 0, stream>>>(beta, pf, shapedirs, posedirs,
                                        v_template, vs, vp);
  k_jreg<<<BATCH * NJOINT, 256, 0, stream>>>(J_regressor, vs, Jb, NJOINT);
  k_kinematics<<<(BATCH + 31) / 32, 32, 0, stream>>>(Rs, Jb, Amat);
  dim3 gs((NVERT + 255) / 256, BATCH);
  k_skin<<<gs, 256, 0, stream>>>(weights, Amat, vp, out);
  k_jreg<<<BATCH * NJREG, 256, 0, stream>>>(joint_regressor, out,
                                            out + (size_t)BATCH * COLS, NJREG);
}